// AdditiveAttention_15101105012754
// MI455X (gfx1250) — compile-verified
//
#include <hip/hip_runtime.h>
#include <hip/hip_bf16.h>
#include <math.h>

// Problem constants from the reference: B=2, H=8, LQ=LK=512, DH=64
#define BH   16
#define LSEQ 512
#define DHD  64

typedef float v2f __attribute__((ext_vector_type(2)));
typedef float v8f __attribute__((ext_vector_type(8)));

__device__ __forceinline__ float htanh_f32(float x) {
#if __has_builtin(__builtin_amdgcn_tanhf)
    return __builtin_amdgcn_tanhf(x);   // V_TANH_F32 (CDNA5 TRANS op)
#else
    return tanhf(x);
#endif
}

// ---------------------------------------------------------------------------
// Kernel 1: projections via V_WMMA_F32_16X16X4_F32.
//   qp[bh][m][e]  = sum_d Q[bh][m][d] * W[e][d]           (e = 0..63 cols of W)
//   kpb[bh][m][e] = sum_d K[bh][m][d] * W[e][64+d] + bias[e]
// One wave computes one 16x16 output tile, accumulating K=64 in 16 steps of 4.
// 2 tensors * 16 bh * (512/16) Mtiles * (64/16) Ntiles = 4096 waves = 512 blocks.
// ---------------------------------------------------------------------------
__global__ __launch_bounds__(256)
void aa_proj_wmma(const float* __restrict__ Q, const float* __restrict__ K,
                  const float* __restrict__ W, const float* __restrict__ bias,
                  float* __restrict__ qp, float* __restrict__ kpb)
{
    const int wave = (int)((blockIdx.x * 256u + threadIdx.x) >> 5);
    const int lane = (int)(threadIdx.x & 31u);

    int t = wave;                    // 0..4095
    const int which = t >> 11;       // 0 = Q-projection, 1 = K-projection
    t &= 2047;
    const int bh = t >> 7;           // 0..15
    t &= 127;
    const int mt = t >> 2;           // 0..31  (16-row tile)
    const int nt = t & 3;            // 0..3   (16-col tile)

    const int half = lane >> 4;      // K sub-block select (A: K+2*half, B: K+2*half)
    const int l    = lane & 15;      // A: row M=l ; B: col N=l ; C/D: col N=l

    const float* src  = (which ? K : Q) + (size_t)bh * (LSEQ * DHD);
    const float* arow = src + (mt * 16 + l) * DHD + 2 * half;        // A[M=l][K..]
    const float* brow = W + (nt * 16 + l) * (2 * DHD) + which * DHD  // B[K..][N=l]
                          + 2 * half;

    v8f c = {};
    #pragma unroll
    for (int s = 0; s < 16; ++s) {
        v2f a = *(const v2f*)(arow + 4 * s);   // A: K = 4s+2*half, 4s+2*half+1
        v2f b = *(const v2f*)(brow + 4 * s);   // B: same K rows, col N=l
        // (neg_a, A, neg_b, B, c_mod, C, reuse_a, reuse_b)
        c = __builtin_amdgcn_wmma_f32_16x16x4_f32(false, a, false, b,
                                                  (short)0, c, false, false);
    }

    float*      dst  = (which ? kpb : qp) + (size_t)bh * (LSEQ * DHD);
    const float badd = which ? bias[nt * 16 + l] : 0.0f;  // bias depends only on e-col
    #pragma unroll
    for (int i = 0; i < 8; ++i) {
        const int row = mt * 16 + half * 8 + i;   // C/D: VGPR i -> M = i (+8 hi half)
        dst[row * DHD + nt * 16 + l] = c[i] + badd;
    }
}

// ---------------------------------------------------------------------------
// Kernel 2: scores[bh][q][k] = sum_e V[e] * tanh(qp[q][e] + kpb[k][e])
// Tile: 16 q x 64 k per block (256 threads). kpb tile staged transposed in LDS
// with stride 65 (bank-conflict-free), qp tile staged row-major (broadcast reads).
// Each thread owns one k column and 4 q rows -> amortizes the k/V loads.
// Grid: 16 bh * 32 qtiles * 8 ktiles = 4096 blocks.
// ---------------------------------------------------------------------------
__global__ __launch_bounds__(256)
void aa_score(const float* __restrict__ qp, const float* __restrict__ kpb,
              const float* __restrict__ Vw, float* __restrict__ out)
{
    __shared__ float qs[16 * DHD];      // [q][e]
    __shared__ float ks[DHD * 65];      // [e][k] padded: bank = (e+k)%64
    __shared__ float vs[DHD];

    int blk = (int)blockIdx.x;
    const int kt = blk & 7;  blk >>= 3;
    const int qt = blk & 31; blk >>= 5;
    const int bh = blk;

    const int   tid  = (int)threadIdx.x;
    const float* qsrc = qp  + ((size_t)bh * LSEQ + qt * 16) * DHD;
    const float* ksrc = kpb + ((size_t)bh * LSEQ + kt * 64) * DHD;

    #pragma unroll
    for (int i = 0; i < 4; ++i) {                 // 16*64 qp tile, coalesced
        const int idx = tid + i * 256;
        qs[idx] = qsrc[idx];
    }
    #pragma unroll
    for (int i = 0; i < 16; ++i) {                // 64*64 kpb tile, transpose in LDS
        const int idx = tid + i * 256;
        ks[(idx & 63) * 65 + (idx >> 6)] = ksrc[idx];
    }
    if (tid < DHD) vs[tid] = Vw[tid];
    __syncthreads();

    const int kl = tid & 63;          // k column (contiguous across quarter-block)
    const int q0 = (tid >> 6) * 4;    // first of 4 q rows

    float acc0 = 0.f, acc1 = 0.f, acc2 = 0.f, acc3 = 0.f;
    #pragma unroll 8
    for (int e = 0; e < DHD; ++e) {
        const float kv = ks[e * 65 + kl];   // conflict-free ds_load
        const float vv = vs[e];             // LDS broadcast
        acc0 += vv * htanh_f32(qs[(q0 + 0) * DHD + e] + kv);
        acc1 += vv * htanh_f32(qs[(q0 + 1) * DHD + e] + kv);
        acc2 += vv * htanh_f32(qs[(q0 + 2) * DHD + e] + kv);
        acc3 += vv * htanh_f32(qs[(q0 + 3) * DHD + e] + kv);
    }

    float* o = out + ((size_t)(bh * LSEQ + qt * 16 + q0)) * LSEQ + kt * 64 + kl;
    o[0 * LSEQ] = acc0;
    o[1 * LSEQ] = acc1;
    o[2 * LSEQ] = acc2;
    o[3 * LSEQ] = acc3;
}

// ---------------------------------------------------------------------------
extern "C" void kernel_launch(void* const* d_in, const int* in_sizes, int n_in,
                              void* d_out, int out_size, void* d_ws, size_t ws_size,
                              hipStream_t stream)
{
    const float* Q  = (const float*)d_in[0];   // (2,8,512,64)
    const float* K  = (const float*)d_in[1];   // (2,8,512,64)
    const float* W  = (const float*)d_in[2];   // (64,128)
    const float* Wb = (const float*)d_in[3];   // (64,)
    const float* Vw = (const float*)d_in[4];   // (64,)
    float*       out = (float*)d_out;          // (2,8,512,512)

    float* qp  = (float*)d_ws;                       // 16*512*64 f32 = 2 MB
    float* kpb = qp + (size_t)BH * LSEQ * DHD;       // another 2 MB

    aa_proj_wmma<<<512, 256, 0, stream>>>(Q, K, W, Wb, qp, kpb);
    aa_score<<<BH * 32 * 8, 256, 0, stream>>>(qp, kpb, Vw, out);
}